// Pointnet_Backbone_16303695856160
// MI455X (gfx1250) — compile-verified
//
#include <hip/hip_runtime.h>
#include <cstddef>

// ---------------------------------------------------------------------------
// CDNA5 (gfx1250) wave32 WMMA f32 16x16x4 helpers
// ---------------------------------------------------------------------------
typedef float v2f __attribute__((ext_vector_type(2)));
typedef float v8f __attribute__((ext_vector_type(8)));

__device__ __forceinline__ v8f wmma_f32_16x16x4(v2f a, v2f b, v8f c) {
  // 8 args: (neg_a, A, neg_b, B, c_mod, C, reuse_a, reuse_b)
  return __builtin_amdgcn_wmma_f32_16x16x4_f32(false, a, false, b, (short)0, c,
                                               false, false);
}

// ---------------------------------------------------------------------------
// 16x16-tiled linear (one wave per output tile) reading A from LDS.
// Y = relu(X * W^T + bias). W is (cout x cin) row-major; cin is a multiple of
// 4 (weights pre-padded with zeros where needed), so all operand loads are
// unconditional 8-byte loads and EXEC stays all-ones through the WMMA loop.
// ---------------------------------------------------------------------------
__device__ __forceinline__ void lds_linear_wmma(
    const float* __restrict__ X, int ldX, const float* __restrict__ W,
    const float* __restrict__ bias, float* __restrict__ Y, int ldY, int rows,
    int cin, int cout) {
  const int lane = threadIdx.x & 31;
  const int wave = threadIdx.x >> 5;
  const int nw = blockDim.x >> 5;
  const int lidx = lane & 15;
  const int half = lane >> 4;
  const int koff = half * 2;
  const int ct = cout >> 4;
  const int tot = (rows >> 4) * ct;
  for (int t = wave; t < tot; t += nw) {
    const int r0 = (t / ct) << 4;
    const int c0 = (t % ct) << 4;
    const int n = c0 + lidx;  // output column handled by this lane
    const float* Xrow = X + (r0 + lidx) * ldX + koff;
    const float* Wrow = W + n * cin + koff;
    v8f acc = {0.f, 0.f, 0.f, 0.f, 0.f, 0.f, 0.f, 0.f};
    for (int kb = 0; kb < cin; kb += 4) {
      const v2f a = *(const v2f*)(Xrow + kb);
      const v2f b = *(const v2f*)(Wrow + kb);
      acc = wmma_f32_16x16x4(a, b, acc);
    }
    const float bb = bias[n];
#pragma unroll
    for (int v = 0; v < 8; ++v) {
      float val = acc[v] + bb;
      val = val > 0.f ? val : 0.f;
      Y[(r0 + v + half * 8) * ldY + n] = val;
    }
  }
}

// ---------------------------------------------------------------------------
// kNN: one wave32 per query. K passes of lane-strided scan + lexicographic
// (dist, idx) min reduction via __shfl_xor. Writes idx (and optionally d2).
// ---------------------------------------------------------------------------
__global__ void knn_kernel(const float* __restrict__ xyz_q,
                           const float* __restrict__ xyz_r, int B, int Nq,
                           int Nr, int strideQ, int strideR, int K,
                           int* __restrict__ idx_out,
                           float* __restrict__ d2_out) {
  const int lane = threadIdx.x & 31;
  const int gw = (blockIdx.x * blockDim.x + threadIdx.x) >> 5;
  if (gw >= B * Nq) return;
  const int b = gw / Nq;
  const int q = gw - b * Nq;
  const float* qp = xyz_q + ((size_t)b * strideQ + q) * 3;
  const float qx = qp[0], qy = qp[1], qz = qp[2];
  const float* rbase = xyz_r + (size_t)b * strideR * 3;

  float lastd = -3.4e38f;
  int lasti = -1;
  for (int s = 0; s < K; ++s) {
    float bd = 3.4e38f;
    int bi = 0x7fffffff;
    for (int j = lane; j < Nr; j += 32) {
      const float dx = rbase[j * 3 + 0] - qx;
      const float dy = rbase[j * 3 + 1] - qy;
      const float dz = rbase[j * 3 + 2] - qz;
      const float d = dx * dx + dy * dy + dz * dz;
      const bool gt = (d > lastd) || (d == lastd && j > lasti);
      if (gt && (d < bd || (d == bd && j < bi))) {
        bd = d;
        bi = j;
      }
    }
#pragma unroll
    for (int off = 16; off > 0; off >>= 1) {
      const float od = __shfl_xor(bd, off, 32);
      const int oi = __shfl_xor(bi, off, 32);
      if (od < bd || (od == bd && oi < bi)) {
        bd = od;
        bi = oi;
      }
    }
    if (lane == 0) {
      idx_out[(size_t)gw * K + s] = bi;
      if (d2_out) d2_out[(size_t)gw * K + s] = bd;
    }
    lastd = bd;
    lasti = bi;
  }
}

// ---------------------------------------------------------------------------
// Fused SA stage: one workgroup per (batch, center). Gather grouped features
// into LDS, run 3 WMMA linear+relu layers (ping-pong LDS), max-pool rows.
// All weight matrices are (cout x cin) with cin a multiple of 4 (padded).
// ---------------------------------------------------------------------------
template <int NS, int C0P, int C1, int C2, int C3, int CMAX>
__global__ void sa_fused_kernel(const float* __restrict__ xyz, int xyzRows,
                                const float* __restrict__ feats, int featRows,
                                const int* __restrict__ idx, int npoint,
                                const float* __restrict__ W0,
                                const float* __restrict__ b0,
                                const float* __restrict__ W1,
                                const float* __restrict__ b1,
                                const float* __restrict__ W2,
                                const float* __restrict__ b2,
                                float* __restrict__ out) {
  __shared__ float bufA[NS * CMAX];
  __shared__ float bufB[NS * CMAX];
  const int b = blockIdx.x / npoint;
  const int p = blockIdx.x - b * npoint;
  const int* myIdx = idx + (size_t)blockIdx.x * NS;

  if (feats == nullptr) {
    // grouped input = xyz[neighbor] - xyz[center], padded 3 -> C0P
    for (int e = threadIdx.x; e < NS * C0P; e += blockDim.x) {
      const int r = e / C0P, c = e - r * C0P;
      float v = 0.f;
      if (c < 3) {
        const int j = myIdx[r];
        v = xyz[((size_t)b * xyzRows + j) * 3 + c] -
            xyz[((size_t)b * xyzRows + p) * 3 + c];
      }
      bufA[r * CMAX + c] = v;
    }
  } else {
    const int Cf = C0P / 2;
    for (int e = threadIdx.x; e < NS * C0P; e += blockDim.x) {
      const int r = e / C0P, c = e - r * C0P;
      const int j = myIdx[r];
      float v;
      if (c < Cf) {
        v = feats[((size_t)b * featRows + j) * Cf + c] -
            feats[((size_t)b * featRows + p) * Cf + c];
      } else {
        v = feats[((size_t)b * featRows + p) * Cf + (c - Cf)];
      }
      bufA[r * CMAX + c] = v;
    }
  }
  __syncthreads();
  lds_linear_wmma(bufA, CMAX, W0, b0, bufB, CMAX, NS, C0P, C1);
  __syncthreads();
  lds_linear_wmma(bufB, CMAX, W1, b1, bufA, CMAX, NS, C1, C2);
  __syncthreads();
  lds_linear_wmma(bufA, CMAX, W2, b2, bufB, CMAX, NS, C2, C3);
  __syncthreads();
  for (int c = threadIdx.x; c < C3; c += blockDim.x) {
    float m = -3.4e38f;
    for (int r = 0; r < NS; ++r) m = fmaxf(m, bufB[r * CMAX + c]);
    out[(size_t)blockIdx.x * C3 + c] = m;
  }
}

// ---------------------------------------------------------------------------
// Standalone WMMA linear over a global-memory A (rows x lda, zero-padded).
// out = [relu](A * W^T + bias) [+ addsrc]; optional transposed store (B,C,N).
// lda is a multiple of 4 and W rows have exactly lda floats (pre-padded).
// ---------------------------------------------------------------------------
__global__ void linear_kernel(const float* __restrict__ A, int lda,
                              const float* __restrict__ W,
                              const float* __restrict__ bias,
                              float* __restrict__ out, int rows, int cout,
                              int relu, const float* __restrict__ addsrc,
                              int transNb) {
  const int lane = threadIdx.x & 31;
  const int gw = (blockIdx.x * blockDim.x + threadIdx.x) >> 5;
  const int nw = (gridDim.x * blockDim.x) >> 5;
  const int lidx = lane & 15;
  const int half = lane >> 4;
  const int koff = half * 2;
  const int ct = cout >> 4;
  const int tot = (rows >> 4) * ct;
  for (int t = gw; t < tot; t += nw) {
    const int r0 = (t / ct) << 4;
    const int c0 = (t % ct) << 4;
    const int n = c0 + lidx;
    const float* Arow = A + (size_t)(r0 + lidx) * lda + koff;
    const float* Wrow = W + (size_t)n * lda + koff;
    v8f acc = {0.f, 0.f, 0.f, 0.f, 0.f, 0.f, 0.f, 0.f};
    for (int kb = 0; kb < lda; kb += 4) {
      const v2f a = *(const v2f*)(Arow + kb);
      const v2f b = *(const v2f*)(Wrow + kb);
      acc = wmma_f32_16x16x4(a, b, acc);
    }
    const float bb = bias[n];
#pragma unroll
    for (int v = 0; v < 8; ++v) {
      const int row = r0 + v + half * 8;
      float val = acc[v] + bb;
      if (relu) val = val > 0.f ? val : 0.f;
      if (addsrc) val += addsrc[(size_t)row * cout + n];
      if (transNb > 0) {
        const int bb_ = row / transNb;
        const int nn = row - bb_ * transNb;
        out[((size_t)bb_ * cout + n) * transNb + nn] = val;
      } else {
        out[(size_t)row * cout + n] = val;
      }
    }
  }
}

// ---------------------------------------------------------------------------
// Zero-pad a (cout x cinR) weight matrix into (cout x cinP).
// ---------------------------------------------------------------------------
__global__ void pad_weights_kernel(const float* __restrict__ W,
                                   float* __restrict__ Wp, int cout, int cinR,
                                   int cinP) {
  const int total = cout * cinP;
  for (int t = blockIdx.x * blockDim.x + threadIdx.x; t < total;
       t += gridDim.x * blockDim.x) {
    const int n = t / cinP;
    const int k = t - n * cinP;
    Wp[t] = (k < cinR) ? W[n * cinR + k] : 0.f;
  }
}

// ---------------------------------------------------------------------------
// 3-NN inverse-distance interpolation + concat (+ zero pad to Ccat cols).
// cat[row, 0:Ck] = sum_j w_j * known[idx_j], cat[row, Ck:Ck+Cs] = skip[row].
// ---------------------------------------------------------------------------
__global__ void interp_cat_kernel(const float* __restrict__ known, int Nk,
                                  int Ck, const float* __restrict__ skip,
                                  int Cs, const int* __restrict__ idx3,
                                  const float* __restrict__ d23, int B, int Nu,
                                  int Ccat, float* __restrict__ cat) {
  const size_t total = (size_t)B * Nu * Ccat;
  for (size_t t = (size_t)blockIdx.x * blockDim.x + threadIdx.x; t < total;
       t += (size_t)gridDim.x * blockDim.x) {
    const int c = (int)(t % Ccat);
    const size_t row = t / Ccat;
    const int b = (int)(row / Nu);
    float v = 0.f;
    if (c < Ck) {
      const int* id = idx3 + row * 3;
      const float* dd = d23 + row * 3;
      float w0 = 1.f / fmaxf(dd[0], 1e-10f);
      float w1 = 1.f / fmaxf(dd[1], 1e-10f);
      float w2 = 1.f / fmaxf(dd[2], 1e-10f);
      const float s = w0 + w1 + w2;
      w0 /= s;
      w1 /= s;
      w2 /= s;
      v = w0 * known[((size_t)b * Nk + id[0]) * Ck + c] +
          w1 * known[((size_t)b * Nk + id[1]) * Ck + c] +
          w2 * known[((size_t)b * Nk + id[2]) * Ck + c];
    } else if (c < Ck + Cs) {
      v = skip[row * Cs + (c - Ck)];
    }
    cat[row * Ccat + c] = v;
  }
}

__global__ void copy_kernel(const float* __restrict__ src,
                            float* __restrict__ dst, int n) {
  for (int i = blockIdx.x * blockDim.x + threadIdx.x; i < n;
       i += gridDim.x * blockDim.x)
    dst[i] = src[i];
}

// ---------------------------------------------------------------------------
// Host orchestration
// ---------------------------------------------------------------------------
struct Lay {
  const float* W;
  const float* b;
};

static inline void launch_linear(const float* A, int lda, const float* W,
                                 const float* bias, float* out, int rows,
                                 int cout, int relu, const float* addsrc,
                                 int transNb, hipStream_t s) {
  const int tiles = (rows / 16) * (cout / 16);
  const int blocks = (tiles + 7) / 8;  // 8 waves per 256-thread block
  linear_kernel<<<blocks, 256, 0, s>>>(A, lda, W, bias, out, rows, cout, relu,
                                       addsrc, transNb);
}

extern "C" void kernel_launch(void* const* d_in, const int* in_sizes, int n_in,
                              void* d_out, int out_size, void* d_ws,
                              size_t ws_size, hipStream_t stream) {
  (void)n_in;
  (void)out_size;
  (void)ws_size;
  const float* pc = (const float*)d_in[0];  // (2, 8192, 3)

  // --- map flattened params (handles insertion-order and sorted-key order) --
  Lay sa[3][3], fpm[3][2], fpi[3][3], convf;
  {
    int cur = 1;
    auto take = [&](Lay* l) {
      l->W = (const float*)d_in[cur++];
      l->b = (const float*)d_in[cur++];
    };
    if (in_sizes[1] == 96) {  // insertion order: sa, fp_main, fp_inte, conv
      for (int i = 0; i < 3; ++i)
        for (int j = 0; j < 3; ++j) take(&sa[i][j]);
      for (int i = 0; i < 3; ++i)
        for (int j = 0; j < 2; ++j) take(&fpm[i][j]);
      for (int i = 0; i < 3; ++i)
        for (int j = 0; j < 3; ++j) take(&fpi[i][j]);
      take(&convf);
    } else {  // sorted keys: conv_final, fp_inte, fp_main, sa
      take(&convf);
      for (int i = 0; i < 3; ++i)
        for (int j = 0; j < 3; ++j) take(&fpi[i][j]);
      for (int i = 0; i < 3; ++i)
        for (int j = 0; j < 2; ++j) take(&fpm[i][j]);
      for (int i = 0; i < 3; ++i)
        for (int j = 0; j < 3; ++j) take(&sa[i][j]);
    }
  }

  const int B = 2, N0 = 8192, N1 = 2048, N2 = 512, N3 = 128;
  const int K1 = 32, K2 = 48, K3 = 48;

  // --- bump-allocate workspace (~21 MB total) ------------------------------
  char* w = (char*)d_ws;
  auto alloc = [&](size_t bytes) -> void* {
    void* p = (void*)w;
    w += (bytes + 255) & ~(size_t)255;
    return p;
  };
  int* idx1 = (int*)alloc((size_t)B * N1 * K1 * 4);
  int* idx2 = (int*)alloc((size_t)B * N2 * K2 * 4);
  int* idx3 = (int*)alloc((size_t)B * N3 * K3 * 4);
  float* feat1 = (float*)alloc((size_t)B * N1 * 32 * 4);
  float* feat2 = (float*)alloc((size_t)B * N2 * 64 * 4);
  float* feat3 = (float*)alloc((size_t)B * N3 * 128 * 4);
  int* idxFP = (int*)alloc((size_t)B * N0 * 3 * 4);
  float* d2FP = (float*)alloc((size_t)B * N0 * 3 * 4);
  float* cat = (float*)alloc((size_t)B * N0 * 68 * 4);  // max 16384 x 68
  float* hA = (float*)alloc((size_t)B * N0 * 64 * 4);
  float* hB = (float*)alloc((size_t)B * N0 * 64 * 4);
  float* mo = (float*)alloc((size_t)B * N0 * 32 * 4);
  float* fo2 = (float*)alloc((size_t)B * N2 * 128 * 4);
  float* fo1 = (float*)alloc((size_t)B * N1 * 64 * 4);
  float* fo0 = (float*)alloc((size_t)B * N0 * 32 * 4);
  // padded weights (cin -> multiple of 4)
  float* wSa1 = (float*)alloc((size_t)32 * 4 * 4);    // 32x3  -> 32x4
  float* wFpm = (float*)alloc((size_t)32 * 68 * 4);   // 32x67 -> 32x68
  float* wFpi = (float*)alloc((size_t)64 * 68 * 4);   // 64x67 -> 64x68

  float* out_xyz = (float*)d_out;
  float* out_feat = out_xyz + (size_t)B * N0 * 3;

  // pre-pad the three non-multiple-of-4 weight matrices
  pad_weights_kernel<<<1, 128, 0, stream>>>(sa[0][0].W, wSa1, 32, 3, 4);
  pad_weights_kernel<<<9, 256, 0, stream>>>(fpm[0][0].W, wFpm, 32, 67, 68);
  pad_weights_kernel<<<17, 256, 0, stream>>>(fpi[0][0].W, wFpi, 64, 67, 68);

  // output 0: l_xyz[0] == input
  copy_kernel<<<192, 256, 0, stream>>>(pc, out_xyz, B * N0 * 3);

  // ---------------- SA1: 8192 -> 2048 centers, 32 nn, MLP 3->32->32->32 ----
  knn_kernel<<<(B * N1 + 7) / 8, 256, 0, stream>>>(pc, pc, B, N1, N0, N0, N0,
                                                   K1, idx1, nullptr);
  sa_fused_kernel<32, 4, 32, 32, 32, 32><<<B * N1, 128, 0, stream>>>(
      pc, N0, nullptr, 0, idx1, N1, wSa1, sa[0][0].b, sa[0][1].W, sa[0][1].b,
      sa[0][2].W, sa[0][2].b, feat1);

  // ---------------- SA2: 2048 -> 512 centers, 48 nn, MLP 64->64->64->64 ----
  knn_kernel<<<(B * N2 + 7) / 8, 256, 0, stream>>>(pc, pc, B, N2, N1, N0, N0,
                                                   K2, idx2, nullptr);
  sa_fused_kernel<48, 64, 64, 64, 64, 64><<<B * N2, 384, 0, stream>>>(
      pc, N0, feat1, N1, idx2, N2, sa[1][0].W, sa[1][0].b, sa[1][1].W,
      sa[1][1].b, sa[1][2].W, sa[1][2].b, feat2);

  // ---------------- SA3: 512 -> 128 centers, 48 nn, MLP 128^4 --------------
  knn_kernel<<<(B * N3 + 7) / 8, 256, 0, stream>>>(pc, pc, B, N3, N2, N0, N0,
                                                   K3, idx3, nullptr);
  sa_fused_kernel<48, 128, 128, 128, 128, 128><<<B * N3, 768, 0, stream>>>(
      pc, N0, feat2, N2, idx3, N3, sa[2][0].W, sa[2][0].b, sa[2][1].W,
      sa[2][1].b, sa[2][2].W, sa[2][2].b, feat3);

  // ---------------- FP2: up 128 -> 512, cat(128 interp + 64 skip) = 192 ----
  knn_kernel<<<(B * N2 + 7) / 8, 256, 0, stream>>>(pc, pc, B, N2, N3, N0, N0,
                                                   3, idxFP, d2FP);
  {
    const size_t tot = (size_t)B * N2 * 192;
    interp_cat_kernel<<<(int)((tot + 255) / 256), 256, 0, stream>>>(
        feat3, N3, 128, feat2, 64, idxFP, d2FP, B, N2, 192, cat);
  }
  launch_linear(cat, 192, fpm[2][0].W, fpm[2][0].b, hA, B * N2, 128, 1,
                nullptr, 0, stream);
  launch_linear(hA, 128, fpm[2][1].W, fpm[2][1].b, mo, B * N2, 128, 1, nullptr,
                0, stream);
  launch_linear(cat, 192, fpi[2][0].W, fpi[2][0].b, hA, B * N2, 128, 1,
                nullptr, 0, stream);
  launch_linear(hA, 128, fpi[2][1].W, fpi[2][1].b, hB, B * N2, 64, 1, nullptr,
                0, stream);
  launch_linear(hB, 64, fpi[2][2].W, fpi[2][2].b, fo2, B * N2, 128, 1, mo, 0,
                stream);

  // ---------------- FP1: up 512 -> 2048, cat(128 + 32) = 160 ---------------
  knn_kernel<<<(B * N1 + 7) / 8, 256, 0, stream>>>(pc, pc, B, N1, N2, N0, N0,
                                                   3, idxFP, d2FP);
  {
    const size_t tot = (size_t)B * N1 * 160;
    interp_cat_kernel<<<(int)((tot + 255) / 256), 256, 0, stream>>>(
        fo2, N2, 128, feat1, 32, idxFP, d2FP, B, N1, 160, cat);
  }
  launch_linear(cat, 160, fpm[1][0].W, fpm[1][0].b, hA, B * N1, 128, 1,
                nullptr, 0, stream);
  launch_linear(hA, 128, fpm[1][1].W, fpm[1][1].b, mo, B * N1, 64, 1, nullptr,
                0, stream);
  launch_linear(cat, 160, fpi[1][0].W, fpi[1][0].b, hA, B * N1, 128, 1,
                nullptr, 0, stream);
  launch_linear(hA, 128, fpi[1][1].W, fpi[1][1].b, hB, B * N1, 64, 1, nullptr,
                0, stream);
  launch_linear(hB, 64, fpi[1][2].W, fpi[1][2].b, fo1, B * N1, 64, 1, mo, 0,
                stream);

  // ---------------- FP0: up 2048 -> 8192, cat(64 + 3) = 67 (pad 68) --------
  knn_kernel<<<(B * N0 + 7) / 8, 256, 0, stream>>>(pc, pc, B, N0, N1, N0, N0,
                                                   3, idxFP, d2FP);
  {
    const size_t tot = (size_t)B * N0 * 68;
    interp_cat_kernel<<<(int)((tot + 255) / 256), 256, 0, stream>>>(
        fo1, N1, 64, pc, 3, idxFP, d2FP, B, N0, 68, cat);
  }
  launch_linear(cat, 68, wFpm, fpm[0][0].b, hA, B * N0, 32, 1, nullptr, 0,
                stream);
  launch_linear(hA, 32, fpm[0][1].W, fpm[0][1].b, mo, B * N0, 32, 1, nullptr,
                0, stream);
  launch_linear(cat, 68, wFpi, fpi[0][0].b, hA, B * N0, 64, 1, nullptr, 0,
                stream);
  launch_linear(hA, 64, fpi[0][1].W, fpi[0][1].b, hB, B * N0, 64, 1, nullptr,
                0, stream);
  launch_linear(hB, 64, fpi[0][2].W, fpi[0][2].b, fo0, B * N0, 32, 1, mo, 0,
                stream);

  // ---------------- final conv (no relu) + transpose to (B, 32, 8192) ------
  launch_linear(fo0, 32, convf.W, convf.b, out_feat, B * N0, 32, 0, nullptr,
                N0, stream);
}